// MegNetBlock_v4_55851754717352
// MI455X (gfx1250) — compile-verified
//
#include <hip/hip_runtime.h>
#include <hip/hip_bf16.h>

typedef __attribute__((ext_vector_type(2))) float v2f;
typedef __attribute__((ext_vector_type(8))) float v8f;

// ---------------------------------------------------------------------------
// fp32 WMMA: D(16x16) = A(16x4) * B(4x16) + C, full fp32 precision.
// ---------------------------------------------------------------------------
static __device__ __forceinline__ v8f wmma4(v2f a, v2f b, v8f c) {
  return __builtin_amdgcn_wmma_f32_16x16x4_f32(false, a, false, b, (short)0, c,
                                               false, false);
}

static __device__ __forceinline__ v2f ld2(const float* p) {
  return *reinterpret_cast<const v2f*>(p);
}

// B-operand layout for one K=4 chunk: lane l (col = l&15, half h = l>>4)
//   reg0 = W[kr][col], reg1 = W[kr+1][col],  kr = 4k + 2h
static __device__ __forceinline__ v2f ldw(const float* W, int stride, int kr,
                                          int col, int kvalid, int colvalid) {
  v2f r;
  r.x = (col < colvalid && kr     < kvalid) ? W[(size_t)kr * stride + col]     : 0.f;
  r.y = (col < colvalid && kr + 1 < kvalid) ? W[(size_t)(kr + 1) * stride + col] : 0.f;
  return r;
}

static __device__ __forceinline__ v8f splat8(float v) {
  v8f d;
#pragma unroll
  for (int i = 0; i < 8; ++i) d[i] = v;
  return d;
}

static __device__ __forceinline__ v8f relu8(v8f d) {
#pragma unroll
  for (int i = 0; i < 8; ++i) d[i] = fmaxf(d[i], 0.f);
  return d;
}

// D-layout (reg r: row r+8h, col l&15) -> A-layout chunks for a 16x16 tile
// via per-wave LDS (pad 18 floats/row, same-wave LDS is in-order).
static __device__ __forceinline__ void xpose16(float* S, int lane, v8f d,
                                               v2f* out /*4 chunks*/) {
  const int h = lane >> 4, m = lane & 15;
  __builtin_amdgcn_wave_barrier();
#pragma unroll
  for (int r = 0; r < 8; ++r) S[(r + 8 * h) * 18 + m] = d[r];
  __builtin_amdgcn_wave_barrier();
#pragma unroll
  for (int j = 0; j < 4; ++j) {
    out[j].x = S[m * 18 + 4 * j + 2 * h];
    out[j].y = S[m * 18 + 4 * j + 2 * h + 1];
  }
  __builtin_amdgcn_wave_barrier();
}

// ---------------------------------------------------------------------------
// Projection: dst[r,0:8] = relu(src[r,0:32] @ W(32x8) + b), rows = 16*tiles.
// ---------------------------------------------------------------------------
__global__ void megnet_proj(const float* __restrict__ src,
                            const float* __restrict__ W,
                            const float* __restrict__ b,
                            float* __restrict__ dst, int tiles) {
  const int lane = threadIdx.x & 31, wv = threadIdx.x >> 5;
  const int t = blockIdx.x * 8 + wv;
  if (t >= tiles) return;
  const int h = lane >> 4, m = lane & 15, n = m;

  v2f wk[8];
#pragma unroll
  for (int k = 0; k < 8; ++k) wk[k] = ldw(W, 8, 4 * k + 2 * h, n, 32, 8);
  const float bv = (n < 8) ? b[n] : 0.f;

  const int base = t << 4;
  v8f d = splat8(bv);
#pragma unroll
  for (int k = 0; k < 8; ++k) {
    v2f a = ld2(src + (size_t)(base + m) * 32 + 4 * k + 2 * h);
    d = wmma4(a, wk[k], d);
  }
  d = relu8(d);
  if (n < 8) {
#pragma unroll
    for (int r = 0; r < 8; ++r)
      dst[(size_t)(base + r + 8 * h) * 8 + n] = d[r];
  }
}

// ---------------------------------------------------------------------------
// Edge kernel: e_b = relu(EA@We+be) fused; X1=[xb[row],xb[col],e_b,ub[g]] (32)
// MLP 32->16->16->8 ; atomics into node-agg + graph-sums ; edge_out fused.
// ---------------------------------------------------------------------------
__global__ void megnet_edge(const float* __restrict__ ea,
                            const int* __restrict__ ei,
                            const int* __restrict__ batch,
                            const float* __restrict__ xb,
                            const float* __restrict__ ub,
                            const float* __restrict__ Web, const float* __restrict__ beb,
                            const float* __restrict__ W1, const float* __restrict__ b1,
                            const float* __restrict__ W2, const float* __restrict__ b2,
                            const float* __restrict__ W3, const float* __restrict__ b3,
                            const float* __restrict__ Wo, const float* __restrict__ bo,
                            float* __restrict__ agg, float* __restrict__ ncnt,
                            float* __restrict__ egs, float* __restrict__ egc,
                            float* __restrict__ eout, int E, int etiles) {
  __shared__ float S[8][16 * 18];
  __shared__ int RI[8][16];
  __shared__ int EB[8][16];
  const int lane = threadIdx.x & 31, wv = threadIdx.x >> 5;
  const int h = lane >> 4, m = lane & 15, n = m;
  float* Sw = S[wv];

  v2f web[8], w1c[8], w2c[4], w3c[4], wo0[2], wo1[2];
#pragma unroll
  for (int k = 0; k < 8; ++k) {
    web[k] = ldw(Web, 8, 4 * k + 2 * h, n, 32, 8);
    w1c[k] = ldw(W1, 16, 4 * k + 2 * h, n, 32, 16);
  }
#pragma unroll
  for (int k = 0; k < 4; ++k) {
    w2c[k] = ldw(W2, 16, 4 * k + 2 * h, n, 16, 16);
    w3c[k] = ldw(W3, 8, 4 * k + 2 * h, n, 16, 8);
  }
#pragma unroll
  for (int k = 0; k < 2; ++k) {
    wo0[k] = ldw(Wo, 32, 4 * k + 2 * h, n, 8, 32);
    wo1[k] = ldw(Wo, 32, 4 * k + 2 * h, 16 + n, 8, 32);
  }
  const float bvb = (n < 8) ? beb[n] : 0.f;
  const float bv1 = b1[n], bv2 = b2[n];
  const float bv3 = (n < 8) ? b3[n] : 0.f;
  const float bvo0 = bo[n], bvo1 = bo[16 + n];

  const int wid = blockIdx.x * 8 + wv, nw = gridDim.x * 8;
  for (int t = wid; t < etiles; t += nw) {
    const int base = t << 4;
    const int e_m = base + m;
    const int row_m = ei[e_m];
    const int col_m = ei[E + e_m];
    const int g_m = batch[row_m];
    if (h == 0) {
      RI[wv][m] = row_m;
      EB[wv][m] = g_m;
      atomicAdd(ncnt + row_m, 1.f);
      atomicAdd(egc + g_m, 1.f);
    }
    // e_b = relu(EA tile @ We + be)
    v8f d = splat8(bvb);
#pragma unroll
    for (int k = 0; k < 8; ++k) {
      v2f a = ld2(ea + (size_t)e_m * 32 + 4 * k + 2 * h);
      d = wmma4(a, web[k], d);
    }
    d = relu8(d);

    v2f X[8], tmp[4];
    xpose16(Sw, lane, d, tmp);
    X[4] = tmp[0];
    X[5] = tmp[1];
    X[0] = ld2(xb + (size_t)row_m * 8 + 2 * h);
    X[1] = ld2(xb + (size_t)row_m * 8 + 4 + 2 * h);
    X[2] = ld2(xb + (size_t)col_m * 8 + 2 * h);
    X[3] = ld2(xb + (size_t)col_m * 8 + 4 + 2 * h);
    X[6] = ld2(ub + (size_t)g_m * 8 + 2 * h);
    X[7] = ld2(ub + (size_t)g_m * 8 + 4 + 2 * h);

    v8f d1 = splat8(bv1);
#pragma unroll
    for (int k = 0; k < 8; ++k) d1 = wmma4(X[k], w1c[k], d1);
    d1 = relu8(d1);
    v2f y[4];
    xpose16(Sw, lane, d1, y);

    v8f d2 = splat8(bv2);
#pragma unroll
    for (int k = 0; k < 4; ++k) d2 = wmma4(y[k], w2c[k], d2);
    d2 = relu8(d2);
    xpose16(Sw, lane, d2, y);

    v8f d3 = splat8(bv3);  // e_msg, no relu
#pragma unroll
    for (int k = 0; k < 4; ++k) d3 = wmma4(y[k], w3c[k], d3);

    __builtin_amdgcn_wave_barrier();
    if (n < 8) {
#pragma unroll
      for (int r = 0; r < 8; ++r) {
        const int idx = r + 8 * h;
        atomicAdd(agg + (size_t)RI[wv][idx] * 8 + n, d3[r]);
        atomicAdd(egs + (size_t)EB[wv][idx] * 8 + n, d3[r]);
      }
    }
    // edge_out = edge_attr + e_msg @ Weo + beo
    xpose16(Sw, lane, d3, y);  // y[0],y[1] = K=8 chunks
    v8f o0 = splat8(bvo0), o1 = splat8(bvo1);
    o0 = wmma4(y[0], wo0[0], o0);
    o0 = wmma4(y[1], wo0[1], o0);
    o1 = wmma4(y[0], wo1[0], o1);
    o1 = wmma4(y[1], wo1[1], o1);
#pragma unroll
    for (int r = 0; r < 8; ++r) {
      const size_t e2 = (size_t)(base + r + 8 * h) * 32;
      eout[e2 + n] = ea[e2 + n] + o0[r];
      eout[e2 + 16 + n] = ea[e2 + 16 + n] + o1[r];
    }
  }
}

// ---------------------------------------------------------------------------
// Node kernel: X=[agg_mean, x_b, u_b[batch]] (24) -> MLP 24->16->16->8 ;
// atomics into per-graph x sums ; x_out fused.
// ---------------------------------------------------------------------------
__global__ void megnet_node(const float* __restrict__ x,
                            const int* __restrict__ batch,
                            const float* __restrict__ xb,
                            const float* __restrict__ ub,
                            const float* __restrict__ agg,
                            const float* __restrict__ ncnt,
                            const float* __restrict__ W1, const float* __restrict__ b1,
                            const float* __restrict__ W2, const float* __restrict__ b2,
                            const float* __restrict__ W3, const float* __restrict__ b3,
                            const float* __restrict__ Wo, const float* __restrict__ bo,
                            float* __restrict__ xgs, float* __restrict__ ngc,
                            float* __restrict__ xout, int ntiles) {
  __shared__ float S[8][16 * 18];
  __shared__ int GI[8][16];
  const int lane = threadIdx.x & 31, wv = threadIdx.x >> 5;
  const int t = blockIdx.x * 8 + wv;
  if (t >= ntiles) return;
  const int h = lane >> 4, m = lane & 15, n = m;
  float* Sw = S[wv];

  v2f w1c[6], w2c[4], w3c[4], wo0[2], wo1[2];
#pragma unroll
  for (int k = 0; k < 6; ++k) w1c[k] = ldw(W1, 16, 4 * k + 2 * h, n, 24, 16);
#pragma unroll
  for (int k = 0; k < 4; ++k) {
    w2c[k] = ldw(W2, 16, 4 * k + 2 * h, n, 16, 16);
    w3c[k] = ldw(W3, 8, 4 * k + 2 * h, n, 16, 8);
  }
#pragma unroll
  for (int k = 0; k < 2; ++k) {
    wo0[k] = ldw(Wo, 32, 4 * k + 2 * h, n, 8, 32);
    wo1[k] = ldw(Wo, 32, 4 * k + 2 * h, 16 + n, 8, 32);
  }
  const float bv1 = b1[n], bv2 = b2[n];
  const float bv3 = (n < 8) ? b3[n] : 0.f;
  const float bvo0 = bo[n], bvo1 = bo[16 + n];

  const int base = t << 4;
  const int nm = base + m;
  const int g_m = batch[nm];
  if (h == 0) {
    GI[wv][m] = g_m;
    atomicAdd(ngc + g_m, 1.f);
  }
  const float rc = 1.f / fmaxf(ncnt[nm], 1.f);

  v2f X[6];
  X[0] = ld2(agg + (size_t)nm * 8 + 2 * h);
  X[1] = ld2(agg + (size_t)nm * 8 + 4 + 2 * h);
  X[0].x *= rc; X[0].y *= rc; X[1].x *= rc; X[1].y *= rc;
  X[2] = ld2(xb + (size_t)nm * 8 + 2 * h);
  X[3] = ld2(xb + (size_t)nm * 8 + 4 + 2 * h);
  X[4] = ld2(ub + (size_t)g_m * 8 + 2 * h);
  X[5] = ld2(ub + (size_t)g_m * 8 + 4 + 2 * h);

  v8f d1 = splat8(bv1);
#pragma unroll
  for (int k = 0; k < 6; ++k) d1 = wmma4(X[k], w1c[k], d1);
  d1 = relu8(d1);
  v2f y[4];
  xpose16(Sw, lane, d1, y);

  v8f d2 = splat8(bv2);
#pragma unroll
  for (int k = 0; k < 4; ++k) d2 = wmma4(y[k], w2c[k], d2);
  d2 = relu8(d2);
  xpose16(Sw, lane, d2, y);

  v8f d3 = splat8(bv3);  // x_msg
#pragma unroll
  for (int k = 0; k < 4; ++k) d3 = wmma4(y[k], w3c[k], d3);

  __builtin_amdgcn_wave_barrier();
  if (n < 8) {
#pragma unroll
    for (int r = 0; r < 8; ++r)
      atomicAdd(xgs + (size_t)GI[wv][r + 8 * h] * 8 + n, d3[r]);
  }
  xpose16(Sw, lane, d3, y);
  v8f o0 = splat8(bvo0), o1 = splat8(bvo1);
  o0 = wmma4(y[0], wo0[0], o0);
  o0 = wmma4(y[1], wo0[1], o0);
  o1 = wmma4(y[0], wo1[0], o1);
  o1 = wmma4(y[1], wo1[1], o1);
#pragma unroll
  for (int r = 0; r < 8; ++r) {
    const size_t n2 = (size_t)(base + r + 8 * h) * 32;
    xout[n2 + n] = x[n2 + n] + o0[r];
    xout[n2 + 16 + n] = x[n2 + 16 + n] + o1[r];
  }
}

// ---------------------------------------------------------------------------
// Global kernel: X=[u_b, mean(x_msg), mean(e_msg)] (24) -> MLP -> u_out.
// ---------------------------------------------------------------------------
__global__ void megnet_global(const float* __restrict__ u,
                              const float* __restrict__ ub,
                              const float* __restrict__ xgs,
                              const float* __restrict__ ngc,
                              const float* __restrict__ egs,
                              const float* __restrict__ egc,
                              const float* __restrict__ W1, const float* __restrict__ b1,
                              const float* __restrict__ W2, const float* __restrict__ b2,
                              const float* __restrict__ W3, const float* __restrict__ b3,
                              const float* __restrict__ Wo, const float* __restrict__ bo,
                              float* __restrict__ uout, int gtiles) {
  __shared__ float S[8][16 * 18];
  const int lane = threadIdx.x & 31, wv = threadIdx.x >> 5;
  const int t = blockIdx.x * 8 + wv;
  if (t >= gtiles) return;
  const int h = lane >> 4, m = lane & 15, n = m;
  float* Sw = S[wv];

  v2f w1c[6], w2c[4], w3c[4], wo0[2], wo1[2];
#pragma unroll
  for (int k = 0; k < 6; ++k) w1c[k] = ldw(W1, 16, 4 * k + 2 * h, n, 24, 16);
#pragma unroll
  for (int k = 0; k < 4; ++k) {
    w2c[k] = ldw(W2, 16, 4 * k + 2 * h, n, 16, 16);
    w3c[k] = ldw(W3, 8, 4 * k + 2 * h, n, 16, 8);
  }
#pragma unroll
  for (int k = 0; k < 2; ++k) {
    wo0[k] = ldw(Wo, 32, 4 * k + 2 * h, n, 8, 32);
    wo1[k] = ldw(Wo, 32, 4 * k + 2 * h, 16 + n, 8, 32);
  }
  const float bv1 = b1[n], bv2 = b2[n];
  const float bv3 = (n < 8) ? b3[n] : 0.f;
  const float bvo0 = bo[n], bvo1 = bo[16 + n];

  const int base = t << 4;
  const int gm = base + m;
  const float rcx = 1.f / fmaxf(ngc[gm], 1.f);
  const float rce = 1.f / fmaxf(egc[gm], 1.f);

  v2f X[6];
  X[0] = ld2(ub + (size_t)gm * 8 + 2 * h);
  X[1] = ld2(ub + (size_t)gm * 8 + 4 + 2 * h);
  X[2] = ld2(xgs + (size_t)gm * 8 + 2 * h);
  X[3] = ld2(xgs + (size_t)gm * 8 + 4 + 2 * h);
  X[2].x *= rcx; X[2].y *= rcx; X[3].x *= rcx; X[3].y *= rcx;
  X[4] = ld2(egs + (size_t)gm * 8 + 2 * h);
  X[5] = ld2(egs + (size_t)gm * 8 + 4 + 2 * h);
  X[4].x *= rce; X[4].y *= rce; X[5].x *= rce; X[5].y *= rce;

  v8f d1 = splat8(bv1);
#pragma unroll
  for (int k = 0; k < 6; ++k) d1 = wmma4(X[k], w1c[k], d1);
  d1 = relu8(d1);
  v2f y[4];
  xpose16(Sw, lane, d1, y);

  v8f d2 = splat8(bv2);
#pragma unroll
  for (int k = 0; k < 4; ++k) d2 = wmma4(y[k], w2c[k], d2);
  d2 = relu8(d2);
  xpose16(Sw, lane, d2, y);

  v8f d3 = splat8(bv3);  // u_msg
#pragma unroll
  for (int k = 0; k < 4; ++k) d3 = wmma4(y[k], w3c[k], d3);

  xpose16(Sw, lane, d3, y);
  v8f o0 = splat8(bvo0), o1 = splat8(bvo1);
  o0 = wmma4(y[0], wo0[0], o0);
  o0 = wmma4(y[1], wo0[1], o0);
  o1 = wmma4(y[0], wo1[0], o1);
  o1 = wmma4(y[1], wo1[1], o1);
#pragma unroll
  for (int r = 0; r < 8; ++r) {
    const size_t g2 = (size_t)(base + r + 8 * h) * 32;
    uout[g2 + n] = u[g2 + n] + o0[r];
    uout[g2 + 16 + n] = u[g2 + 16 + n] + o1[r];
  }
}

// ---------------------------------------------------------------------------
extern "C" void kernel_launch(void* const* d_in, const int* in_sizes, int n_in,
                              void* d_out, int out_size, void* d_ws, size_t ws_size,
                              hipStream_t stream) {
  const float* x = (const float*)d_in[0];
  const int* ei = (const int*)d_in[1];
  const float* ea = (const float*)d_in[2];
  const float* u = (const float*)d_in[3];
  const int* batch = (const int*)d_in[4];
  // params in setup_inputs insertion order
  const float* Wed = (const float*)d_in[5];  const float* bed = (const float*)d_in[6];
  const float* Wnd = (const float*)d_in[7];  const float* bnd = (const float*)d_in[8];
  const float* Wgd = (const float*)d_in[9];  const float* bgd = (const float*)d_in[10];
  const float* We1 = (const float*)d_in[11]; const float* be1 = (const float*)d_in[12];
  const float* We2 = (const float*)d_in[13]; const float* be2 = (const float*)d_in[14];
  const float* We3 = (const float*)d_in[15]; const float* be3 = (const float*)d_in[16];
  const float* Wn1 = (const float*)d_in[17]; const float* bn1 = (const float*)d_in[18];
  const float* Wn2 = (const float*)d_in[19]; const float* bn2 = (const float*)d_in[20];
  const float* Wn3 = (const float*)d_in[21]; const float* bn3 = (const float*)d_in[22];
  const float* Wg1 = (const float*)d_in[23]; const float* bg1 = (const float*)d_in[24];
  const float* Wg2 = (const float*)d_in[25]; const float* bg2 = (const float*)d_in[26];
  const float* Wg3 = (const float*)d_in[27]; const float* bg3 = (const float*)d_in[28];
  const float* Weo = (const float*)d_in[29]; const float* beo = (const float*)d_in[30];
  const float* Wno = (const float*)d_in[31]; const float* bno = (const float*)d_in[32];
  const float* Wuo = (const float*)d_in[33]; const float* buo = (const float*)d_in[34];

  const int N = in_sizes[0] / 32;
  const int E = in_sizes[2] / 32;
  const int G = in_sizes[3] / 32;
  const int ntiles = N / 16, etiles = E / 16, gtiles = G / 16;

  // workspace layout (floats)
  float* ws = (float*)d_ws;
  float* xb   = ws;
  float* ubuf = xb + (size_t)N * 8;
  float* agg  = ubuf + (size_t)G * 8;   // zeroed region starts here
  float* ncnt = agg + (size_t)N * 8;
  float* egs  = ncnt + (size_t)N;
  float* egc  = egs + (size_t)G * 8;
  float* xgs  = egc + (size_t)G;
  float* ngc  = xgs + (size_t)G * 8;
  const size_t zfloats = (size_t)N * 8 + N + (size_t)G * 8 + G + (size_t)G * 8 + G;
  hipMemsetAsync(agg, 0, zfloats * sizeof(float), stream);

  float* out = (float*)d_out;
  float* xout = out;
  float* eout = out + (size_t)N * 32;
  float* uout = eout + (size_t)E * 32;

  megnet_proj<<<(ntiles + 7) / 8, 256, 0, stream>>>(x, Wnd, bnd, xb, ntiles);
  megnet_proj<<<(gtiles + 7) / 8, 256, 0, stream>>>(u, Wgd, bgd, ubuf, gtiles);

  megnet_edge<<<2048, 256, 0, stream>>>(ea, ei, batch, xb, ubuf,
                                        Wed, bed, We1, be1, We2, be2, We3, be3,
                                        Weo, beo, agg, ncnt, egs, egc, eout,
                                        E, etiles);

  megnet_node<<<(ntiles + 7) / 8, 256, 0, stream>>>(x, batch, xb, ubuf, agg, ncnt,
                                                    Wn1, bn1, Wn2, bn2, Wn3, bn3,
                                                    Wno, bno, xgs, ngc, xout, ntiles);

  megnet_global<<<(gtiles + 7) / 8, 256, 0, stream>>>(u, ubuf, xgs, ngc, egs, egc,
                                                      Wg1, bg1, Wg2, bg2, Wg3, bg3,
                                                      Wuo, buo, uout, gtiles);
}